// GCN_23888608100806
// MI455X (gfx1250) — compile-verified
//
#include <hip/hip_runtime.h>
#include <hip/hip_bf16.h>

// ---------------------------------------------------------------------------
// GCN (3x GCNConv, HID=64) for MI455X / gfx1250, fp32 via V_WMMA_F32_16X16X4_F32
// ---------------------------------------------------------------------------

#define NN      100000          // N_NODES
#define EE      1200000         // N_EDGES
#define HID     64
#define NG      512             // N_GRAPHS
#define NC      2               // NUM_CLASSES

typedef float v2f __attribute__((ext_vector_type(2)));
typedef float v8f __attribute__((ext_vector_type(8)));

// ---------------- utility ----------------
__global__ void k_zero(float* __restrict__ p, long n) {
    long i = (long)blockIdx.x * blockDim.x + threadIdx.x;
    long stride = (long)gridDim.x * blockDim.x;
    for (; i < n; i += stride) p[i] = 0.0f;
}

// ---------------- embedding gather: h[i,:] = emb[x[i],:] ----------------
__global__ void k_gather_emb(const int* __restrict__ x,
                             const float* __restrict__ emb,
                             float* __restrict__ h, int n) {
    int gid = blockIdx.x * blockDim.x + threadIdx.x;   // n*16 threads, float4 each
    if (gid >= n * 16) return;
    int i = gid >> 4, q = gid & 15;
    const float4* src = (const float4*)(emb + (size_t)x[i] * HID);
    float4* dst = (float4*)(h + (size_t)i * HID);
    dst[q] = src[q];
}

// ---------------- degree / mask precompute ----------------
__global__ void k_deg_count(const int* __restrict__ col, float* __restrict__ deg, int e) {
    int i = blockIdx.x * blockDim.x + threadIdx.x;
    if (i < e) atomicAdd(&deg[col[i]], 1.0f);
}
__global__ void k_mask_set(const int* __restrict__ row, const int* __restrict__ col,
                           float* __restrict__ mask, int e) {
    int i = blockIdx.x * blockDim.x + threadIdx.x;
    if (i < e) { mask[row[i]] = 1.0f; mask[col[i]] = 1.0f; }
}
__global__ void k_deg_fin(const float* __restrict__ deg,
                          float* __restrict__ dinv, float* __restrict__ drec, int n) {
    int i = blockIdx.x * blockDim.x + threadIdx.x;
    if (i >= n) return;
    float d = deg[i] + 1.0f;                  // + self loop
    dinv[i] = 1.0f / sqrtf(d);
    drec[i] = 1.0f / d;
}

// ---------------- fp32 WMMA GEMM: out[N,64] = h[N,64] @ W[64,64] ----------------
// One wave computes a 16x16 output tile; K=64 => 16 x V_WMMA_F32_16X16X4_F32.
// A 16x4 layout  : lanes 0-15 -> {K=0,K=1} in vgpr {0,1}; lanes 16-31 -> {K=2,K=3}
// B 4x16 layout  : symmetric (N = lane%16 within a VGPR, K pair split on lane half)
// C/D 16x16      : vgpr r -> M = r + 8*(lane/16), N = lane%16
__global__ __launch_bounds__(256) void k_gemm_wmma(const float* __restrict__ h,
                                                   const float* __restrict__ W,
                                                   float* __restrict__ out, int nrows) {
    int wave  = (blockIdx.x * blockDim.x + threadIdx.x) >> 5;
    int lane  = threadIdx.x & 31;
    int tileN = wave & 3;                 // 64 cols / 16
    int tileM = wave >> 2;
    if (tileM * 16 >= nrows) return;      // wave-uniform: EXEC stays all-1s for WMMA
    int half = lane >> 4;
    int l16  = lane & 15;

    const float* arow = h + (size_t)(tileM * 16 + l16) * HID;
    const float* bcol = W + tileN * 16 + l16;

    v8f c = {};
#pragma unroll
    for (int k0 = 0; k0 < HID; k0 += 4) {
        int ka = k0 + 2 * half;
        v2f a; a.x = arow[ka];               a.y = arow[ka + 1];
        v2f b; b.x = bcol[(size_t)ka * HID]; b.y = bcol[(size_t)(ka + 1) * HID];
        c = __builtin_amdgcn_wmma_f32_16x16x4_f32(
                /*neg_a=*/false, a, /*neg_b=*/false, b,
                /*c_mod=*/(short)0, c, /*reuse_a=*/false, /*reuse_b=*/false);
    }

    float* orow = out + (size_t)(tileM * 16 + 8 * half) * HID + tileN * 16 + l16;
#pragma unroll
    for (int r = 0; r < 8; ++r) orow[(size_t)r * HID] = c[r];
}

// ---------------- per-edge gather/scale/scatter-add ----------------
// 16 threads per edge, float4 per thread; atomics land in L2 (agg fits in 192MB L2)
__global__ __launch_bounds__(256) void k_edge_scatter(const int* __restrict__ row,
                                                      const int* __restrict__ col,
                                                      const float* __restrict__ hW,
                                                      const float* __restrict__ dinv,
                                                      float* __restrict__ agg, int e) {
    long gid = (long)blockIdx.x * blockDim.x + threadIdx.x;
    if (gid >= (long)e * 16) return;
    int ei = (int)(gid >> 4), q = (int)(gid & 15);
    int r = row[ei], cn = col[ei];
    float norm = dinv[r] * dinv[cn];
    float4 v = ((const float4*)(hW + (size_t)r * HID))[q];
    float* dst = agg + (size_t)cn * HID + q * 4;
    atomicAdd(dst + 0, v.x * norm);
    atomicAdd(dst + 1, v.y * norm);
    atomicAdd(dst + 2, v.z * norm);
    atomicAdd(dst + 3, v.w * norm);
}

// ---------------- combine: h = [relu](agg + hW/deg + b) ----------------
__global__ void k_combine(const float* __restrict__ agg, const float* __restrict__ hW,
                          const float* __restrict__ drec, const float* __restrict__ b,
                          float* __restrict__ hout, int n, int do_relu) {
    long gid = (long)blockIdx.x * blockDim.x + threadIdx.x;
    if (gid >= (long)n * HID) return;
    int i = (int)(gid >> 6), f = (int)(gid & 63);
    float v = agg[gid] + hW[gid] * drec[i] + b[f];
    hout[gid] = do_relu ? fmaxf(v, 0.0f) : v;
}

// ---------------- masked mean-pool scatter ----------------
__global__ void k_pool(const float* __restrict__ h, const float* __restrict__ mask,
                       const int* __restrict__ batch,
                       float* __restrict__ sums, float* __restrict__ cnts, int n) {
    long gid = (long)blockIdx.x * blockDim.x + threadIdx.x;
    if (gid >= (long)n * 16) return;
    int i = (int)(gid >> 4), q = (int)(gid & 15);
    float m = mask[i];
    if (m == 0.0f) return;
    int g = batch[i];
    float4 v = ((const float4*)(h + (size_t)i * HID))[q];
    float* dst = sums + (size_t)g * HID + q * 4;
    atomicAdd(dst + 0, v.x);
    atomicAdd(dst + 1, v.y);
    atomicAdd(dst + 2, v.z);
    atomicAdd(dst + 3, v.w);
    if (q == 0) atomicAdd(&cnts[g], 1.0f);
}

// ---------------- final linear head: out[G,2] ----------------
__global__ void k_final(const float* __restrict__ sums, const float* __restrict__ cnts,
                        const float* __restrict__ lW, const float* __restrict__ lb,
                        float* __restrict__ out) {
    int gid = blockIdx.x * blockDim.x + threadIdx.x;
    if (gid >= NG * NC) return;
    int g = gid >> 1, c = gid & 1;
    float inv = 1.0f / fmaxf(cnts[g], 1.0f);
    float acc = lb[c];
#pragma unroll
    for (int k = 0; k < HID; ++k)
        acc += sums[(size_t)g * HID + k] * inv * lW[k * NC + c];
    out[gid] = acc;
}

// ---------------------------------------------------------------------------
extern "C" void kernel_launch(void* const* d_in, const int* in_sizes, int n_in,
                              void* d_out, int out_size, void* d_ws, size_t ws_size,
                              hipStream_t stream) {
    const int*   x    = (const int*)d_in[0];
    const int*   row  = (const int*)d_in[1];            // edge_index[0,:]
    const int*   col  = (const int*)d_in[1] + EE;       // edge_index[1,:]
    // d_in[2] = edge_type (unused by reference math)
    const int*   batch = (const int*)d_in[3];
    const float* emb   = (const float*)d_in[4];
    const float* Ws[3] = { (const float*)d_in[5], (const float*)d_in[7], (const float*)d_in[9]  };
    const float* bs[3] = { (const float*)d_in[6], (const float*)d_in[8], (const float*)d_in[10] };
    const float* linW  = (const float*)d_in[11];
    const float* linb  = (const float*)d_in[12];
    float* out = (float*)d_out;

    // workspace layout (floats)
    float* ws   = (float*)d_ws;
    float* deg  = ws;                    // NN
    float* mask = deg  + NN;             // NN
    float* sums = mask + NN;             // NG*HID
    float* cnts = sums + (size_t)NG*HID; // NG
    float* dinv = cnts + NG;             // NN
    float* drec = dinv + NN;             // NN
    float* h    = drec + NN;             // NN*HID
    float* hW   = h    + (size_t)NN*HID; // NN*HID
    float* agg  = hW   + (size_t)NN*HID; // NN*HID

    const int B = 256;
    // zero accumulators: deg, mask, sums, cnts are contiguous
    long zn = 2L*NN + (long)NG*HID + NG;
    k_zero<<<1024, B, 0, stream>>>(deg, zn);

    // embedding gather
    k_gather_emb<<<(NN*16 + B - 1)/B, B, 0, stream>>>(x, emb, h, NN);

    // degree + mask + norms
    k_deg_count<<<(EE + B - 1)/B, B, 0, stream>>>(col, deg, EE);
    k_mask_set <<<(EE + B - 1)/B, B, 0, stream>>>(row, col, mask, EE);
    k_deg_fin  <<<(NN + B - 1)/B, B, 0, stream>>>(deg, dinv, drec, NN);

    // three GCN layers (relu on first two)
    const int gemm_waves  = (NN/16) * 4;                  // 25000 waves
    const int gemm_blocks = gemm_waves / (B/32);          // 3125 blocks
    for (int l = 0; l < 3; ++l) {
        k_gemm_wmma<<<gemm_blocks, B, 0, stream>>>(h, Ws[l], hW, NN);
        k_zero<<<1024, B, 0, stream>>>(agg, (long)NN*HID);
        long nsct = (long)EE * 16;
        k_edge_scatter<<<(int)((nsct + B - 1)/B), B, 0, stream>>>(row, col, hW, dinv, agg, EE);
        k_combine<<<(int)(((long)NN*HID + B - 1)/B), B, 0, stream>>>(
            agg, hW, drec, bs[l], h, NN, l < 2 ? 1 : 0);
    }

    // pooled mean over masked nodes, then linear head
    k_pool<<<(NN*16 + B - 1)/B, B, 0, stream>>>(h, mask, batch, sums, cnts, NN);
    k_final<<<(NG*NC + B - 1)/B, B, 0, stream>>>(sums, cnts, linW, linb, out);
}